// UniversalGNN_8340826488870
// MI455X (gfx1250) — compile-verified
//
#include <hip/hip_runtime.h>

#define N_NODES   50000
#define N_EDGES   800000
#define FDIM      128
#define N_CLASSES 16
#define N_GRAPHS  64
#define LDS_STRIDE 132   // 128 + 4 pad floats: spreads row-strided reads over all 64 LDS banks

typedef __attribute__((ext_vector_type(2))) float v2f;
typedef __attribute__((ext_vector_type(8))) float v8f;

// Pointer types matching __builtin_amdgcn_global_load_async_to_lds_b128's
// signature exactly: gcc vector_size(16) int, AS1 (global) / AS3 (LDS).
typedef int v4i_vs __attribute__((vector_size(16)));
typedef __attribute__((address_space(1))) v4i_vs gbuf_t;
typedef __attribute__((address_space(3))) v4i_vs sbuf_t;

// ---------------------------------------------------------------------------
// Degree / normalization prep
// ---------------------------------------------------------------------------
__global__ void deg_init_kernel(float* __restrict__ deg) {
    int i = blockIdx.x * blockDim.x + threadIdx.x;
    if (i < N_NODES) deg[i] = 1.0f;  // self-loop contributes 1
}

__global__ void deg_acc_kernel(const int* __restrict__ dst, float* __restrict__ deg) {
    int e = blockIdx.x * blockDim.x + threadIdx.x;
    if (e < N_EDGES) atomicAdd(&deg[dst[e]], 1.0f);
}

__global__ void deg_rsqrt_kernel(float* __restrict__ deg) {
    int i = blockIdx.x * blockDim.x + threadIdx.x;
    if (i < N_NODES) deg[i] = rsqrtf(deg[i]);  // deg >= 1 always
}

// ---------------------------------------------------------------------------
// Fully-fp32 WMMA GEMM:  C[N,128] = A[N,128] @ W[128,128]
// One wave per 16x16 output tile, 8 waves/block cover all 8 column tiles.
// The shared 16x128 A tile (8KB) is staged once per block into LDS via the
// gfx1250 async-copy path (GLOBAL_LOAD_ASYNC_TO_LDS_B128 + s_wait_asynccnt),
// removing the 8x duplicate global reads the direct-load version incurred.
// V_WMMA_F32_16X16X4_F32, K chained 128/4 = 32 accumulating steps.
// A layout (16x4 f32): lanes 0-15 -> K=0(v0),K=1(v1); lanes 16-31 -> K=2,K=3.
// D layout: lane&15 = col, VGPR r -> row = r + 8*(lane>>4).
// ---------------------------------------------------------------------------
__global__ __launch_bounds__(256) void gemm_wmma_f32_kernel(
    const float* __restrict__ A, const float* __restrict__ W,
    float* __restrict__ C) {
    __shared__ float As[16 * LDS_STRIDE];

    const int t    = threadIdx.x;
    const int row0 = blockIdx.x * 16;  // N divisible by 16 (50000 = 3125*16)

    // --- stage 16x128 A tile into padded LDS: 512 x 16B chunks ---
#if __has_builtin(__builtin_amdgcn_global_load_async_to_lds_b128)
    {
        const float* gbase = A + (size_t)row0 * FDIM;
#pragma unroll
        for (int i = 0; i < 2; ++i) {
            int c = t + i * 256;
            int r = c >> 5;        // tile row
            int q = (c & 31) * 4;  // float offset within row
            __builtin_amdgcn_global_load_async_to_lds_b128(
                (gbuf_t*)(gbase + r * FDIM + q),
                (sbuf_t*)(As + r * LDS_STRIDE + q),
                0, 0);
        }
    }
#if __has_builtin(__builtin_amdgcn_s_wait_asynccnt)
    __builtin_amdgcn_s_wait_asynccnt(0);
#else
    asm volatile("s_wait_asynccnt 0" ::: "memory");
#endif
#else
    {
        const float4* g4 = (const float4*)(A + (size_t)row0 * FDIM);
#pragma unroll
        for (int i = 0; i < 2; ++i) {
            int c = t + i * 256;
            int r = c >> 5;
            int q = (c & 31) * 4;
            *(float4*)(As + r * LDS_STRIDE + q) = g4[c];
        }
    }
#endif
    __syncthreads();

    const int wave = t >> 5;
    const int lane = t & 31;
    const int half = lane >> 4;   // 0: lanes 0-15, 1: lanes 16-31
    const int l15  = lane & 15;
    const int col0 = wave * 16;
    const int koff = half * 2;

    const float* arp = As + l15 * LDS_STRIDE;

    v8f acc = {};
#pragma unroll
    for (int k = 0; k < FDIM; k += 4) {
        v2f a, b;
        a.x = arp[k + koff];
        a.y = arp[k + koff + 1];
        b.x = W[(k + koff)     * FDIM + col0 + l15];
        b.y = W[(k + koff + 1) * FDIM + col0 + l15];
        acc = __builtin_amdgcn_wmma_f32_16x16x4_f32(
            /*neg_a=*/false, a, /*neg_b=*/false, b,
            /*c_mod=*/(short)0, acc, /*reuse_a=*/false, /*reuse_b=*/false);
    }

#pragma unroll
    for (int r = 0; r < 8; ++r) {
        C[(long long)(row0 + r + 8 * half) * FDIM + col0 + l15] = acc[r];
    }
}

// ---------------------------------------------------------------------------
// Zero a float buffer (float4-wide)
// ---------------------------------------------------------------------------
__global__ void zero4_kernel(float4* __restrict__ p, int n4) {
    int i = blockIdx.x * blockDim.x + threadIdx.x;
    if (i < n4) p[i] = make_float4(0.f, 0.f, 0.f, 0.f);
}

__global__ void zero_kernel(float* __restrict__ p, int n) {
    int i = blockIdx.x * blockDim.x + threadIdx.x;
    if (i < n) p[i] = 0.f;
}

// ---------------------------------------------------------------------------
// Edge scatter: out[dst] += H[src] * dinv[src]*dinv[dst]
// One edge per 32 lanes, float4 per lane -> coalesced 512B gather per edge,
// 4 f32 atomics per lane into L2 (working set is L2-resident, 25.6MB << 192MB)
// ---------------------------------------------------------------------------
__global__ void scatter_kernel(const float* __restrict__ H, float* __restrict__ out,
                               const int* __restrict__ src, const int* __restrict__ dst,
                               const float* __restrict__ dinv) {
    long long idx = (long long)blockIdx.x * blockDim.x + threadIdx.x;
    int e = (int)(idx >> 5);
    int d = ((int)idx & 31) * 4;
    if (e >= N_EDGES) return;
    int s = src[e], t = dst[e];
    float nrm = dinv[s] * dinv[t];
    const float4 v = *(const float4*)(H + (long long)s * FDIM + d);
    float* o = out + (long long)t * FDIM + d;
    atomicAdd(o + 0, v.x * nrm);
    atomicAdd(o + 1, v.y * nrm);
    atomicAdd(o + 2, v.z * nrm);
    atomicAdd(o + 3, v.w * nrm);
}

// ---------------------------------------------------------------------------
// Fuse self-loop term + bias (+ optional ReLU):
//   agg[i,d] = act( agg[i,d] + H[i,d]*dinv[i]^2 + b[d] )
// ---------------------------------------------------------------------------
__global__ void finish_kernel(float* __restrict__ agg, const float* __restrict__ H,
                              const float* __restrict__ dinv,
                              const float* __restrict__ bias, int relu) {
    long long i = (long long)blockIdx.x * blockDim.x + threadIdx.x;
    if (i >= (long long)N_NODES * FDIM) return;
    int node = (int)(i >> 7);
    int d    = (int)(i & 127);
    float di = dinv[node];
    float v = agg[i] + H[i] * di * di + bias[d];
    if (relu) v = fmaxf(v, 0.f);
    agg[i] = v;
}

// ---------------------------------------------------------------------------
// Scatter-mean pooling: sums[g,:] += h[i,:], cnt[g] += 1
// ---------------------------------------------------------------------------
__global__ void pool_kernel(const float* __restrict__ H, const int* __restrict__ batch,
                            float* __restrict__ sums, float* __restrict__ cnt) {
    long long i = (long long)blockIdx.x * blockDim.x + threadIdx.x;
    if (i >= (long long)N_NODES * FDIM) return;
    int node = (int)(i >> 7);
    int d    = (int)(i & 127);
    int g = batch[node];
    atomicAdd(&sums[g * FDIM + d], H[i]);
    if (d == 0) atomicAdd(&cnt[g], 1.0f);
}

// ---------------------------------------------------------------------------
// Classifier: out[g,c] = (sums[g,:]/max(cnt[g],1)) . Wc[:,c] + bc[c]
// 64*16 = 1024 outputs -> one block
// ---------------------------------------------------------------------------
__global__ void cls_kernel(const float* __restrict__ sums, const float* __restrict__ cnt,
                           const float* __restrict__ Wc, const float* __restrict__ bc,
                           float* __restrict__ out) {
    int g = threadIdx.x >> 4;   // blockDim.x == 1024
    int c = threadIdx.x & 15;
    float inv = 1.0f / fmaxf(cnt[g], 1.0f);
    float acc = bc[c];
#pragma unroll 8
    for (int k = 0; k < FDIM; ++k)
        acc += sums[g * FDIM + k] * inv * Wc[k * N_CLASSES + c];
    out[g * N_CLASSES + c] = acc;
}

// ---------------------------------------------------------------------------
extern "C" void kernel_launch(void* const* d_in, const int* in_sizes, int n_in,
                              void* d_out, int out_size, void* d_ws, size_t ws_size,
                              hipStream_t stream) {
    const float* x     = (const float*)d_in[0];
    const int*   ei    = (const int*)  d_in[1];
    const int*   batch = (const int*)  d_in[2];
    const float* W0 = (const float*)d_in[3];
    const float* b0 = (const float*)d_in[4];
    const float* W1 = (const float*)d_in[5];
    const float* b1 = (const float*)d_in[6];
    const float* W2 = (const float*)d_in[7];
    const float* b2 = (const float*)d_in[8];
    const float* Wc = (const float*)d_in[9];
    const float* bc = (const float*)d_in[10];
    float* out = (float*)d_out;

    const int* src = ei;            // edge_index[0,:]
    const int* dst = ei + N_EDGES;  // edge_index[1,:]

    // workspace layout
    char*  ws  = (char*)d_ws;
    size_t off = 0;
    float* dinv = (float*)(ws + off); off += (((size_t)N_NODES * 4) + 255) & ~(size_t)255;
    float* hA   = (float*)(ws + off); off += (size_t)N_NODES * FDIM * 4;
    float* hB   = (float*)(ws + off); off += (size_t)N_NODES * FDIM * 4;
    float* sums = (float*)(ws + off); off += (size_t)N_GRAPHS * FDIM * 4;
    float* cnt  = (float*)(ws + off); off += (size_t)N_GRAPHS * 4;
    (void)ws_size; (void)in_sizes; (void)n_in; (void)out_size;

    const int B = 256;
    const int gridNode  = (N_NODES + B - 1) / B;
    const int gridEdge  = (N_EDGES + B - 1) / B;
    const int gridFeat  = (N_NODES * FDIM + B - 1) / B;         // 25000
    const int gridFeat4 = (N_NODES * FDIM / 4 + B - 1) / B;     // 6250
    const int gridScat  = (int)(((long long)N_EDGES * 32 + B - 1) / B); // 100000
    const int gridGemm  = N_NODES / 16;                         // 3125

    // --- degree / normalization ---
    deg_init_kernel<<<gridNode, B, 0, stream>>>(dinv);
    deg_acc_kernel<<<gridEdge, B, 0, stream>>>(dst, dinv);
    deg_rsqrt_kernel<<<gridNode, B, 0, stream>>>(dinv);

    // --- layer 0 ---
    gemm_wmma_f32_kernel<<<gridGemm, B, 0, stream>>>(x, W0, hA);
    zero4_kernel<<<gridFeat4, B, 0, stream>>>((float4*)hB, N_NODES * FDIM / 4);
    scatter_kernel<<<gridScat, B, 0, stream>>>(hA, hB, src, dst, dinv);
    finish_kernel<<<gridFeat, B, 0, stream>>>(hB, hA, dinv, b0, 1);

    // --- layer 1 ---
    gemm_wmma_f32_kernel<<<gridGemm, B, 0, stream>>>(hB, W1, hA);
    zero4_kernel<<<gridFeat4, B, 0, stream>>>((float4*)hB, N_NODES * FDIM / 4);
    scatter_kernel<<<gridScat, B, 0, stream>>>(hA, hB, src, dst, dinv);
    finish_kernel<<<gridFeat, B, 0, stream>>>(hB, hA, dinv, b1, 1);

    // --- layer 2 (no ReLU) ---
    gemm_wmma_f32_kernel<<<gridGemm, B, 0, stream>>>(hB, W2, hA);
    zero4_kernel<<<gridFeat4, B, 0, stream>>>((float4*)hB, N_NODES * FDIM / 4);
    scatter_kernel<<<gridScat, B, 0, stream>>>(hA, hB, src, dst, dinv);
    finish_kernel<<<gridFeat, B, 0, stream>>>(hB, hA, dinv, b2, 0);

    // --- pooling + classifier ---
    zero_kernel<<<(N_GRAPHS * FDIM + N_GRAPHS + B - 1) / B, B, 0, stream>>>(
        sums, N_GRAPHS * FDIM + N_GRAPHS);  // sums and cnt are contiguous
    pool_kernel<<<gridFeat, B, 0, stream>>>(hB, batch, sums, cnt);
    cls_kernel<<<1, 1024, 0, stream>>>(sums, cnt, Wc, bc, out);
}